// AttnDecoderRNN_new_18966575579384
// MI455X (gfx1250) — compile-verified
//
#include <hip/hip_runtime.h>
#include <math.h>

typedef float v2f __attribute__((ext_vector_type(2)));
typedef float v8f __attribute__((ext_vector_type(8)));

#define HH 2048
#define VV 8192
#define MLL 4096
#define LL 64

// ---------------------------------------------------------------------------
// WMMA-based GEMV: partial[s*M + row] = sum_{k in chunk s} W[row][k] * x[k]
// Uses V_WMMA_F32_16X16X4_F32. Layout assumptions (CDNA5 ISA 7.12.2):
//   A(16x4 f32):  lane l holds A[m=l&15][k=2*(l>>4)+reg]  (documented)
//   B(4x16 f32):  reg r: lanes0-15 K=r, lanes16-31 K=r+2 (C/D striping analogy)
//                 -> with x broadcast across N, lane l reg r = x[kb+2*(l>>4)+r]
//   D(16x16 f32): reg r: lanes0-15 M=r, lanes16-31 M=r+8 (documented)
// Each wave owns a 16-row tile; block = 4 waves = 64 rows. x chunk staged in LDS.
// ---------------------------------------------------------------------------
__global__ __launch_bounds__(128)
void gemv_wmma_kernel(const float* __restrict__ Wm, const float* __restrict__ x,
                      float* __restrict__ part, int M, int K, int KCH) {
  __shared__ float xs[2048];
  const int tid  = threadIdx.x;
  const int wave = tid >> 5;
  const int lane = tid & 31;
  const int m    = lane & 15;
  const int h    = lane >> 4;
  const int koff = blockIdx.y * KCH;

  for (int i = tid; i < KCH; i += 128) xs[i] = x[koff + i];
  __syncthreads();

  const int row0 = blockIdx.x * 64 + wave * 16;
  const float* wp = Wm + (size_t)(row0 + m) * (size_t)K + koff + 2 * h;
  const float* bp = xs + 2 * h;

  v8f acc = {0.f, 0.f, 0.f, 0.f, 0.f, 0.f, 0.f, 0.f};
  for (int kb = 0; kb < KCH; kb += 16) {
    v2f a0 = *(const v2f*)(wp + kb);
    v2f a1 = *(const v2f*)(wp + kb + 4);
    v2f a2 = *(const v2f*)(wp + kb + 8);
    v2f a3 = *(const v2f*)(wp + kb + 12);
    v2f b0 = *(const v2f*)(bp + kb);
    v2f b1 = *(const v2f*)(bp + kb + 4);
    v2f b2 = *(const v2f*)(bp + kb + 8);
    v2f b3 = *(const v2f*)(bp + kb + 12);
    __builtin_prefetch(wp + kb + 256, 0, 0);  // global_prefetch_b8 ahead of A stream
    acc = __builtin_amdgcn_wmma_f32_16x16x4_f32(false, a0, false, b0, (short)0, acc, false, false);
    acc = __builtin_amdgcn_wmma_f32_16x16x4_f32(false, a1, false, b1, (short)0, acc, false, false);
    acc = __builtin_amdgcn_wmma_f32_16x16x4_f32(false, a2, false, b2, (short)0, acc, false, false);
    acc = __builtin_amdgcn_wmma_f32_16x16x4_f32(false, a3, false, b3, (short)0, acc, false, false);
  }
  // D: reg r -> row (8*h + r) at N==0 lanes (lane 0 and lane 16)
  if (m == 0) {
    float* yp = part + (size_t)blockIdx.y * (size_t)M + row0 + 8 * h;
#pragma unroll
    for (int r = 0; r < 8; ++r) yp[r] = acc[r];
  }
}

// y = act(sum_s partial[s] + bias)
__global__ void reduce_bias_kernel(const float* __restrict__ part, const float* __restrict__ bias,
                                   float* __restrict__ y, int M, int S, int act) {
  int i = blockIdx.x * 256 + threadIdx.x;
  if (i >= M) return;
  float s = 0.f;
  for (int j = 0; j < S; ++j) s += part[(size_t)j * (size_t)M + i];
  if (bias) s += bias[i];
  if (act == 1) s = fmaxf(s, 0.f);
  y[i] = s;
}

// attn_applied: partial[s*N + n] = sum_{k in chunk} w[k] * E[k][n]   (E row-major KxN)
__global__ __launch_bounds__(256)
void colgemv_kernel(const float* __restrict__ w, const float* __restrict__ E,
                    float* __restrict__ part, int N, int KCH) {
  int col  = blockIdx.x * 256 + threadIdx.x;
  int koff = blockIdx.y * KCH;
  const float* ep = E + (size_t)koff * (size_t)N + col;
  float acc = 0.f;
  for (int k = 0; k < KCH; k += 4) {
    float w0 = w[koff + k], w1 = w[koff + k + 1], w2 = w[koff + k + 2], w3 = w[koff + k + 3];
    acc += w0 * ep[(size_t)k * N];
    acc += w1 * ep[(size_t)(k + 1) * N];
    acc += w2 * ep[(size_t)(k + 2) * N];
    acc += w3 * ep[(size_t)(k + 3) * N];
  }
  part[(size_t)blockIdx.y * (size_t)N + col] = acc;
}

// embed = mean over 64 gathered rows; build cat1=[embed,h0], cat2[0:H]=embed
__global__ void embed_cat_kernel(const int* __restrict__ toks, const float* __restrict__ emb,
                                 const float* __restrict__ h0,
                                 float* __restrict__ cat1, float* __restrict__ cat2) {
  int i = blockIdx.x * 256 + threadIdx.x;  // 0..2*H-1
  if (i < HH) {
    float s = 0.f;
    for (int t = 0; t < LL; ++t) s += emb[(size_t)toks[t] * HH + i];
    float e = s * (1.0f / 64.0f);
    cat1[i] = e;
    cat2[i] = e;
  } else {
    int j = i - HH;
    cat1[HH + j] = h0[j];
  }
}

__global__ __launch_bounds__(1024)
void softmax4096_kernel(const float* __restrict__ in, float* __restrict__ out) {
  __shared__ float red[1024];
  int tid = threadIdx.x;
  float v[4];
  float vmax = -1e30f;
  for (int j = 0; j < 4; ++j) { v[j] = in[tid + j * 1024]; vmax = fmaxf(vmax, v[j]); }
  red[tid] = vmax; __syncthreads();
  for (int s = 512; s > 0; s >>= 1) { if (tid < s) red[tid] = fmaxf(red[tid], red[tid + s]); __syncthreads(); }
  float mx = red[0]; __syncthreads();
  float lsum = 0.f;
  for (int j = 0; j < 4; ++j) { v[j] = __expf(v[j] - mx); lsum += v[j]; }
  red[tid] = lsum; __syncthreads();
  for (int s = 512; s > 0; s >>= 1) { if (tid < s) red[tid] += red[tid + s]; __syncthreads(); }
  float inv = 1.0f / red[0];
  for (int j = 0; j < 4; ++j) out[tid + j * 1024] = v[j] * inv;
}

__global__ void gru_kernel(const float* __restrict__ gi, const float* __restrict__ gh,
                           const float* __restrict__ h0, float* __restrict__ hnew) {
  int i = blockIdx.x * 256 + threadIdx.x;
  if (i >= HH) return;
  float r = 1.f / (1.f + __expf(-(gi[i] + gh[i])));
  float z = 1.f / (1.f + __expf(-(gi[HH + i] + gh[HH + i])));
  float n = tanhf(gi[2 * HH + i] + r * gh[2 * HH + i]);
  hnew[i] = (1.f - z) * n + z * h0[i];
}

// t = lin*(1-res*sig(c5p)) + hist*sig(c5p); out = softmax(t)
__global__ __launch_bounds__(1024)
void final_softmax_kernel(const float* __restrict__ lin, const float* __restrict__ hist,
                          const float* __restrict__ c5p, float* __restrict__ out) {
  __shared__ float tbuf[8192];
  __shared__ float red[1024];
  int tid = threadIdx.x;
  float vmax = -1e30f;
  for (int j = 0; j < 8; ++j) {
    int i = tid + j * 1024;
    float hv  = hist[i];
    float val = 1.f / (1.f + __expf(-c5p[i]));
    float res = (hv != 0.f) ? 1.f : 0.f;
    float t   = lin[i] * (1.f - res * val) + hv * val;
    tbuf[i] = t;
    vmax = fmaxf(vmax, t);
  }
  red[tid] = vmax; __syncthreads();
  for (int s = 512; s > 0; s >>= 1) { if (tid < s) red[tid] = fmaxf(red[tid], red[tid + s]); __syncthreads(); }
  float mx = red[0]; __syncthreads();
  float lsum = 0.f;
  for (int j = 0; j < 8; ++j) { int i = tid + j * 1024; float e = __expf(tbuf[i] - mx); tbuf[i] = e; lsum += e; }
  red[tid] = lsum; __syncthreads();
  for (int s = 512; s > 0; s >>= 1) { if (tid < s) red[tid] += red[tid + s]; __syncthreads(); }
  float inv = 1.f / red[0];
  for (int j = 0; j < 8; ++j) { int i = tid + j * 1024; out[i] = tbuf[i] * inv; }
}

extern "C" void kernel_launch(void* const* d_in, const int* in_sizes, int n_in,
                              void* d_out, int out_size, void* d_ws, size_t ws_size,
                              hipStream_t stream) {
  const int*   toks   = (const int*)d_in[0];
  const float* hidden = (const float*)d_in[1];
  const float* enc    = (const float*)d_in[2];
  const float* hist   = (const float*)d_in[3];
  /* d_in[4] last_hidden: unused by reference */
  const float* emb    = (const float*)d_in[5];
  const float* attn_W = (const float*)d_in[6];
  const float* attn_b = (const float*)d_in[7];
  const float* comb_W = (const float*)d_in[8];
  const float* comb_b = (const float*)d_in[9];
  const float* gWih   = (const float*)d_in[10];
  const float* gWhh   = (const float*)d_in[11];
  const float* gbih   = (const float*)d_in[12];
  const float* gbhh   = (const float*)d_in[13];
  const float* out_W  = (const float*)d_in[14];
  const float* out_b  = (const float*)d_in[15];
  const float* c5_W   = (const float*)d_in[16];
  const float* c5_b   = (const float*)d_in[17];

  float* out      = (float*)d_out;
  float* o_output = out;          // 8192
  float* o_hnew   = out + 8192;   // 2048
  float* o_attnw  = out + 10240;  // 4096

  float* ws     = (float*)d_ws;
  float* cat1   = ws;             // 4096  [embed, h0]
  float* cat2   = ws + 4096;      // 4096  [embed, attn_applied]
  float* logits = ws + 8192;      // 4096
  float* xact   = ws + 12288;     // 2048
  float* gi     = ws + 14336;     // 6144
  float* gh     = ws + 20480;     // 6144
  float* lin    = ws + 26624;     // 8192
  float* c5p    = ws + 34816;     // 8192
  float* p_c5   = ws + 43008;     // 4*8192
  float* p_attn = ws + 75776;     // 2*4096
  float* p_app  = ws + 83968;     // 4*2048
  float* p_comb = ws + 92160;     // 2*2048
  float* p_gih  = ws + 96256;     // 6144
  float* p_ghh  = ws + 102400;    // 6144
  float* p_out  = ws + 108544;    // 8192

  // Independent branch: c5_pre = c5_W @ hist + c5_b   (256 MB stream, split-K=4)
  gemv_wmma_kernel<<<dim3(VV / 64, 4), 128, 0, stream>>>(c5_W, hist, p_c5, VV, VV, 2048);
  reduce_bias_kernel<<<VV / 256, 256, 0, stream>>>(p_c5, c5_b, c5p, VV, 4, 0);

  // embed + concat(embed, h0)
  embed_cat_kernel<<<(2 * HH) / 256, 256, 0, stream>>>(toks, emb, hidden, cat1, cat2);

  // attn_logits = attn_W @ cat1 + attn_b ; softmax -> attn_weights (output)
  gemv_wmma_kernel<<<dim3(MLL / 64, 2), 128, 0, stream>>>(attn_W, cat1, p_attn, MLL, 2 * HH, 2048);
  reduce_bias_kernel<<<MLL / 256, 256, 0, stream>>>(p_attn, attn_b, logits, MLL, 2, 0);
  softmax4096_kernel<<<1, 1024, 0, stream>>>(logits, o_attnw);

  // attn_applied = attn_weights @ encoder_outputs -> cat2[H:]
  colgemv_kernel<<<dim3(HH / 256, 4), 256, 0, stream>>>(o_attnw, enc, p_app, HH, 1024);
  reduce_bias_kernel<<<HH / 256, 256, 0, stream>>>(p_app, nullptr, cat2 + HH, HH, 4, 0);

  // x = relu(comb_W @ cat2 + comb_b)
  gemv_wmma_kernel<<<dim3(HH / 64, 2), 128, 0, stream>>>(comb_W, cat2, p_comb, HH, 2 * HH, 2048);
  reduce_bias_kernel<<<HH / 256, 256, 0, stream>>>(p_comb, comb_b, xact, HH, 2, 1);

  // GRU: gi = Wih @ x + bih ; gh = Whh @ h0 + bhh ; gates -> h_new (output)
  gemv_wmma_kernel<<<dim3((3 * HH) / 64, 1), 128, 0, stream>>>(gWih, xact, p_gih, 3 * HH, HH, 2048);
  reduce_bias_kernel<<<(3 * HH) / 256, 256, 0, stream>>>(p_gih, gbih, gi, 3 * HH, 1, 0);
  gemv_wmma_kernel<<<dim3((3 * HH) / 64, 1), 128, 0, stream>>>(gWhh, hidden, p_ghh, 3 * HH, HH, 2048);
  reduce_bias_kernel<<<(3 * HH) / 256, 256, 0, stream>>>(p_ghh, gbhh, gh, 3 * HH, 1, 0);
  gru_kernel<<<HH / 256, 256, 0, stream>>>(gi, gh, hidden, o_hnew);

  // lin = out_W @ h_new + out_b
  gemv_wmma_kernel<<<dim3(VV / 64, 1), 128, 0, stream>>>(out_W, o_hnew, p_out, VV, HH, 2048);
  reduce_bias_kernel<<<VV / 256, 256, 0, stream>>>(p_out, out_b, lin, VV, 1, 0);

  // fused history-gated softmax -> output
  final_softmax_kernel<<<1, 1024, 0, stream>>>(lin, hist, c5p, o_output);
}